// EquiFormerEnc_83451214561664
// MI455X (gfx1250) — compile-verified
//
#include <hip/hip_runtime.h>
#include <hip/hip_bf16.h>
#include <math.h>

// ---------------------------------------------------------------------------
// CDNA5 (gfx1250) fused EquiFormer encoder, fp32 end-to-end using
// V_WMMA_F32_16X16X4_F32 for every GEMM tile.
// ---------------------------------------------------------------------------

typedef float v8f __attribute__((ext_vector_type(8)));
typedef float v2f __attribute__((ext_vector_type(2)));

__device__ __forceinline__ v8f wmma_f32(v2f a, v2f b, v8f c) {
  // 8-arg pattern: (neg_a, A, neg_b, B, c_mod, C, reuse_a, reuse_b)
  return __builtin_amdgcn_wmma_f32_16x16x4_f32(false, a, false, b, (short)0, c,
                                               false, false);
}

__device__ __forceinline__ float sigmoidf_(float x) {
  return 1.0f / (1.0f + __expf(-x));
}
__device__ __forceinline__ float siluf_(float x) { return x * sigmoidf_(x); }

// ---------------------------------------------------------------------------
// Kernel 1: initial embedding. emb0[b,p,0,:] = feat @ W0 + b0 ; m=1..8 zeroed
// pc layout: (16, 4096, 6); emb0: (16, 4096, 9, 32)
// ---------------------------------------------------------------------------
__global__ void init_embed(const float* __restrict__ pc,
                           const float* __restrict__ W0,
                           const float* __restrict__ b0,
                           float* __restrict__ emb0) {
  int p = blockIdx.x * blockDim.x + threadIdx.x;
  if (p >= 16 * 4096) return;
  const float* f = pc + (size_t)p * 6 + 3;
  float f0 = f[0], f1 = f[1], f2 = f[2];
  float* e = emb0 + (size_t)p * 9 * 32;
#pragma unroll 4
  for (int c = 0; c < 32; ++c)
    e[c] = b0[c] + f0 * W0[c] + f1 * W0[32 + c] + f2 * W0[64 + c];
  for (int i = 32; i < 9 * 32; ++i) e[i] = 0.0f;
}

// ---------------------------------------------------------------------------
// Kernel 2: exact 16-NN per destination point.
// Source point s (s in [0,Ns)) lives at original index s*stride.
// Dest point nd lives at original index nd*4*stride (or origin when last!=0).
// Stable ascending insertion == jax top_k(-d2) tie order (lower index first).
// ---------------------------------------------------------------------------
__global__ void knn_kernel(const float* __restrict__ pc, int* __restrict__ nbr,
                           int Ns, int Nd, int stride, int last) {
  int tid = blockIdx.x * blockDim.x + threadIdx.x;
  if (tid >= 16 * Nd) return;
  int b = tid / Nd, nd = tid - b * Nd;
  float dcx = 0.f, dcy = 0.f, dcz = 0.f;
  if (!last) {
    const float* dp = pc + ((size_t)b * 4096 + (size_t)nd * 4 * stride) * 6;
    dcx = dp[0]; dcy = dp[1]; dcz = dp[2];
  }
  float bd[16];
  int bi[16];
#pragma unroll
  for (int q = 0; q < 16; ++q) { bd[q] = 1e30f; bi[q] = 0; }
  const float* base = pc + (size_t)b * 4096 * 6;
  for (int j = 0; j < Ns; ++j) {
    const float* sp = base + (size_t)j * stride * 6;
    float dx = sp[0] - dcx, dy = sp[1] - dcy, dz = sp[2] - dcz;
    float d2 = dx * dx + dy * dy + dz * dz;
    if (d2 < bd[15]) {
      float nv = d2; int nj = j;
#pragma unroll
      for (int q = 0; q < 16; ++q) {
        if (nv < bd[q]) {
          float tv = bd[q]; int ti = bi[q];
          bd[q] = nv; bi[q] = nj;
          nv = tv; nj = ti;
        }
      }
    }
  }
#pragma unroll
  for (int q = 0; q < 16; ++q) nbr[(size_t)tid * 16 + q] = bi[q];
}

// ---------------------------------------------------------------------------
// Kernel 3 (per scale): fully fused attention block for one (batch, dst point)
// per 128-thread workgroup (4 wave32s). HCV == EC at every scale.
// emb_in : (16, Ns, 9, CI)   emb_out : (16, Nd, 9, CO)
// ---------------------------------------------------------------------------
template <int CI, int EC, int CO, int FF>
__global__ __launch_bounds__(128) void scale_kernel(
    const float* __restrict__ pc, const float* __restrict__ emb_in,
    float* __restrict__ emb_out, const int* __restrict__ nbr,
    const float* __restrict__ wr1, const float* __restrict__ br1,
    const float* __restrict__ wr2, const float* __restrict__ br2,
    const float* __restrict__ wa, const float* __restrict__ wg,
    const float* __restrict__ bg, const float* __restrict__ wv,
    const float* __restrict__ wo, const float* __restrict__ wf1,
    const float* __restrict__ wf2, int Ns, int Nd, int stride, float cutoff,
    int last) {
  constexpr int HCV = EC;      // H*cv == EC at every scale
  constexpr int CV = EC / 4;   // per-head value width

  __shared__ float s_rbf[16][64];
  __shared__ float s_h[16][EC];
  __shared__ float s_e[16][EC];
  __shared__ float s_gate[16][HCV];
  __shared__ float s_msg[16][HCV];  // rows 9..15 zero-padded
  __shared__ float s_o[16][CO];
  __shared__ float s_f[16][FF];
  __shared__ float s_logit[64];
  __shared__ float s_alpha[64];
  __shared__ int s_nb[16];
  __shared__ float s_d[16];

  const int bn = blockIdx.x;  // b*Nd + nd
  const int b = bn / Nd, nd = bn - b * Nd;
  const int t = threadIdx.x;
  const int lane = t & 31, wave = t >> 5;
  const int half = lane >> 4;  // hi/lo 16-lane group
  const int l16 = lane & 15;

  if (t < 16) s_nb[t] = nbr[(size_t)bn * 16 + t];
  __syncthreads();

  // --- edge lengths -------------------------------------------------------
  float dcx = 0.f, dcy = 0.f, dcz = 0.f;
  if (!last) {
    const float* dp = pc + ((size_t)b * 4096 + (size_t)nd * 4 * stride) * 6;
    dcx = dp[0]; dcy = dp[1]; dcz = dp[2];
  }
  if (t < 16) {
    const float* sp = pc + ((size_t)b * 4096 + (size_t)s_nb[t] * stride) * 6;
    float ex = sp[0] - dcx, ey = sp[1] - dcy, ez = sp[2] - dcz;
    s_d[t] = sqrtf(ex * ex + ey * ey + ez * ez + 1e-12f);
  }
  __syncthreads();

  // --- RBF basis 16x64 ----------------------------------------------------
  {
    const float invstd = 64.0f / cutoff;      // std = cutoff/NB
    const float mustep = cutoff / 63.0f;      // linspace(0,cutoff,64)
    const float invcut = 1.0f / cutoff;
    for (int i = t; i < 16 * 64; i += 128) {
      int k = i >> 6, q = i & 63;
      float d = s_d[k];
      float z = (d - (float)q * mustep) * invstd;
      float u = fminf(fmaxf(d * invcut, 0.0f), 1.0f);
      float env = 0.5f * (__cosf(3.14159265358979f * u) + 1.0f);
      s_rbf[k][q] = __expf(-0.5f * z * z) * env;
    }
  }
  __syncthreads();

  // --- h = silu(rbf @ wr1 + br1)  (16 x EC, K=64) -------------------------
  for (int nt = wave; nt < EC / 16; nt += 4) {
    v8f acc = {};
    for (int kk = 0; kk < 64; kk += 4) {
      int ka = kk + 2 * half;
      v2f a = {s_rbf[l16][ka], s_rbf[l16][ka + 1]};
      v2f bb = {wr1[(size_t)ka * EC + nt * 16 + l16],
                wr1[(size_t)(ka + 1) * EC + nt * 16 + l16]};
      acc = wmma_f32(a, bb, acc);
    }
    float bias = br1[nt * 16 + l16];
#pragma unroll
    for (int r = 0; r < 8; ++r)
      s_h[r + 8 * half][nt * 16 + l16] = siluf_(acc[r] + bias);
  }
  __syncthreads();

  // --- e = h @ wr2 + br2  (16 x EC, K=EC) ---------------------------------
  for (int nt = wave; nt < EC / 16; nt += 4) {
    v8f acc = {};
    for (int kk = 0; kk < EC; kk += 4) {
      int ka = kk + 2 * half;
      v2f a = {s_h[l16][ka], s_h[l16][ka + 1]};
      v2f bb = {wr2[(size_t)ka * EC + nt * 16 + l16],
                wr2[(size_t)(ka + 1) * EC + nt * 16 + l16]};
      acc = wmma_f32(a, bb, acc);
    }
    float bias = br2[nt * 16 + l16];
#pragma unroll
    for (int r = 0; r < 8; ++r)
      s_e[r + 8 * half][nt * 16 + l16] = acc[r] + bias;
  }
  __syncthreads();

  // --- gate = sigmoid(e @ wg + bg)  (16 x HCV, K=EC) ----------------------
  for (int nt = wave; nt < HCV / 16; nt += 4) {
    v8f acc = {};
    for (int kk = 0; kk < EC; kk += 4) {
      int ka = kk + 2 * half;
      v2f a = {s_e[l16][ka], s_e[l16][ka + 1]};
      v2f bb = {wg[(size_t)ka * HCV + nt * 16 + l16],
                wg[(size_t)(ka + 1) * HCV + nt * 16 + l16]};
      acc = wmma_f32(a, bb, acc);
    }
    float bias = bg[nt * 16 + l16];
#pragma unroll
    for (int r = 0; r < 8; ++r)
      s_gate[r + 8 * half][nt * 16 + l16] = sigmoidf_(acc[r] + bias);
  }
  __syncthreads();

  // --- attention logits + softmax over the 16 neighbors -------------------
  if (t < 64) {
    int k = t >> 2, h = t & 3;
    const float* se = emb_in + (((size_t)b * Ns + s_nb[k]) * 9 + 0) * CI;
    float acc = 0.0f;
    for (int c = 0; c < CI; ++c) acc += se[c] * wa[c * 4 + h];
    if (!last) {
      const float* de = emb_in + (((size_t)b * Ns + (size_t)nd * 4) * 9) * CI;
      for (int c = 0; c < CI; ++c) acc += de[c] * wa[(CI + c) * 4 + h];
    }
    for (int c = 0; c < EC; ++c) acc += s_e[k][c] * wa[(2 * CI + c) * 4 + h];
    s_logit[k * 4 + h] = acc;
  }
  __syncthreads();
  if (t < 4) {
    float mx = -1e30f;
    for (int k = 0; k < 16; ++k) mx = fmaxf(mx, s_logit[k * 4 + t]);
    float ex[16], sum = 0.0f;
    for (int k = 0; k < 16; ++k) {
      ex[k] = __expf(s_logit[k * 4 + t] - mx);
      sum += ex[k];
    }
    float inv = 1.0f / sum;
    for (int k = 0; k < 16; ++k) s_alpha[k * 4 + t] = ex[k] * inv;
  }
  // zero-pad msg rows 9..15 while the softmax lanes work
  for (int i = t; i < 7 * HCV; i += 128) s_msg[9 + i / HCV][i % HCV] = 0.0f;
  __syncthreads();

  // --- msg[m,:] = sum_k alpha*gate * (src_e[k,m,:] @ wv)  -----------------
  // A fragments gathered straight from global (whole emb fits in 192MB L2).
  constexpr int NT6 = HCV / 16;
  for (int tt = wave; tt < 9 * NT6; tt += 4) {
    int m = tt / NT6, nt = tt - m * NT6;
    const float* Abase =
        emb_in + (((size_t)b * Ns + s_nb[l16]) * 9 + m) * CI;  // row = nbr l16
    v8f acc = {};
    for (int kk = 0; kk < CI; kk += 4) {
      int ka = kk + 2 * half;
      v2f a = {Abase[ka], Abase[ka + 1]};
      v2f bb = {wv[(size_t)ka * HCV + nt * 16 + l16],
                wv[(size_t)(ka + 1) * HCV + nt * 16 + l16]};
      acc = wmma_f32(a, bb, acc);
    }
    int col = nt * 16 + l16;
    int hh = col / CV;
    float part = 0.0f;
#pragma unroll
    for (int r = 0; r < 8; ++r) {
      int krow = r + 8 * half;  // neighbor index for this acc element
      part += acc[r] * s_alpha[krow * 4 + hh] * s_gate[krow][col];
    }
    part += __shfl_xor(part, 16, 32);  // combine k rows 0..7 with 8..15
    if (half == 0) s_msg[m][col] = part;
  }
  __syncthreads();

  // --- o = msg_pad(16 x HCV) @ wo  (16 x CO) ------------------------------
  for (int nt = wave; nt < CO / 16; nt += 4) {
    v8f acc = {};
    for (int kk = 0; kk < HCV; kk += 4) {
      int ka = kk + 2 * half;
      v2f a = {s_msg[l16][ka], s_msg[l16][ka + 1]};
      v2f bb = {wo[(size_t)ka * CO + nt * 16 + l16],
                wo[(size_t)(ka + 1) * CO + nt * 16 + l16]};
      acc = wmma_f32(a, bb, acc);
    }
#pragma unroll
    for (int r = 0; r < 8; ++r) s_o[r + 8 * half][nt * 16 + l16] = acc[r];
  }
  __syncthreads();

  // --- f = silu(o @ wf1)  (16 x FF, K=CO) ---------------------------------
  for (int nt = wave; nt < FF / 16; nt += 4) {
    v8f acc = {};
    for (int kk = 0; kk < CO; kk += 4) {
      int ka = kk + 2 * half;
      v2f a = {s_o[l16][ka], s_o[l16][ka + 1]};
      v2f bb = {wf1[(size_t)ka * FF + nt * 16 + l16],
                wf1[(size_t)(ka + 1) * FF + nt * 16 + l16]};
      acc = wmma_f32(a, bb, acc);
    }
#pragma unroll
    for (int r = 0; r < 8; ++r) s_f[r + 8 * half][nt * 16 + l16] = siluf_(acc[r]);
  }
  __syncthreads();

  // --- out = o + f @ wf2, write rows 0..8 ---------------------------------
  for (int nt = wave; nt < CO / 16; nt += 4) {
    v8f acc = {};
    for (int kk = 0; kk < FF; kk += 4) {
      int ka = kk + 2 * half;
      v2f a = {s_f[l16][ka], s_f[l16][ka + 1]};
      v2f bb = {wf2[(size_t)ka * CO + nt * 16 + l16],
                wf2[(size_t)(ka + 1) * CO + nt * 16 + l16]};
      acc = wmma_f32(a, bb, acc);
    }
    int col = nt * 16 + l16;
#pragma unroll
    for (int r = 0; r < 8; ++r) {
      int row = r + 8 * half;
      if (row < 9) {
        emb_out[(((size_t)b * Nd + nd) * 9 + row) * CO + col] =
            acc[r] + s_o[row][col];
      }
    }
  }
}

// ---------------------------------------------------------------------------
// Kernel 4: per-degree-block RMS norm + gain.  emb4: (16, 9, 64) -> out same.
// ---------------------------------------------------------------------------
__global__ void final_norm(const float* __restrict__ emb,
                           const float* __restrict__ g,
                           float* __restrict__ out) {
  __shared__ float red[64];
  int b = blockIdx.x / 3, l = blockIdx.x % 3;
  int m0 = (l == 0) ? 0 : ((l == 1) ? 1 : 4);
  int m1 = (l == 0) ? 1 : ((l == 1) ? 4 : 9);
  int c = threadIdx.x;
  float s = 0.0f;
  for (int m = m0; m < m1; ++m) {
    float x = emb[((size_t)b * 9 + m) * 64 + c];
    s += x * x;
  }
  red[c] = s;
  __syncthreads();
  if (c == 0) {
    float tot = 0.0f;
    for (int i = 0; i < 64; ++i) tot += red[i];
    red[0] = rsqrtf(tot / (float)((m1 - m0) * 64) + 1e-6f);
  }
  __syncthreads();
  float inv = red[0];
  float gn = g[l * 64 + c];
  for (int m = m0; m < m1; ++m) {
    float x = emb[((size_t)b * 9 + m) * 64 + c];
    out[((size_t)b * 9 + m) * 64 + c] = x * inv * gn;
  }
}

// ---------------------------------------------------------------------------
// Host side
// ---------------------------------------------------------------------------
extern "C" void kernel_launch(void* const* d_in, const int* in_sizes, int n_in,
                              void* d_out, int out_size, void* d_ws,
                              size_t ws_size, hipStream_t stream) {
  // d_in layout = setup_inputs() dict insertion order, params flattened
  // recursively in insertion order:
  // 0:point_cloud 1:quat 2:qpos 3:epos 4:W0 5:b0 6:g_norm
  // then per scale n (base 7+11n): wr1,br1,wr2,br2,wa,wg,bg,wv,wo,wf1,wf2
  const float* pc = (const float*)d_in[0];
  const float* W0 = (const float*)d_in[4];
  const float* b0 = (const float*)d_in[5];
  const float* gN = (const float*)d_in[6];
  auto S = [&](int n, int j) { return (const float*)d_in[7 + n * 11 + j]; };

  // workspace: ping-pong emb buffers + nbr scratch
  char* ws = (char*)d_ws;
  float* bufA = (float*)ws;                               // <= 75.5 MB
  float* bufB = (float*)(ws + (size_t)76 * 1024 * 1024);  // <= 37.8 MB
  int* nbr = (int*)(ws + (size_t)114 * 1024 * 1024);      // 1 MB

  // initial embedding into bufA (emb0: 16 x 4096 x 9 x 32)
  init_embed<<<(16 * 4096 + 127) / 128, 128, 0, stream>>>(pc, W0, b0, bufA);

  // scale 0: 4096 -> 1024, CI=32 EC=16 CO=64 FF=32, cutoff=0.05*0.99
  knn_kernel<<<(16 * 1024 + 127) / 128, 128, 0, stream>>>(pc, nbr, 4096, 1024,
                                                          1, 0);
  scale_kernel<32, 16, 64, 32><<<16 * 1024, 128, 0, stream>>>(
      pc, bufA, bufB, nbr, S(0, 0), S(0, 1), S(0, 2), S(0, 3), S(0, 4),
      S(0, 5), S(0, 6), S(0, 7), S(0, 8), S(0, 9), S(0, 10), 4096, 1024, 1,
      0.05f * 0.99f, 0);

  // scale 1: 1024 -> 256, CI=64 EC=32 CO=128 FF=64, cutoff=0.2*0.99
  knn_kernel<<<(16 * 256 + 127) / 128, 128, 0, stream>>>(pc, nbr, 1024, 256, 4,
                                                         0);
  scale_kernel<64, 32, 128, 64><<<16 * 256, 128, 0, stream>>>(
      pc, bufB, bufA, nbr, S(1, 0), S(1, 1), S(1, 2), S(1, 3), S(1, 4),
      S(1, 5), S(1, 6), S(1, 7), S(1, 8), S(1, 9), S(1, 10), 1024, 256, 4,
      0.2f * 0.99f, 0);

  // scale 2: 256 -> 64, CI=128 EC=64 CO=128 FF=128, cutoff=0.8*0.99
  knn_kernel<<<(16 * 64 + 127) / 128, 128, 0, stream>>>(pc, nbr, 256, 64, 16,
                                                        0);
  scale_kernel<128, 64, 128, 128><<<16 * 64, 128, 0, stream>>>(
      pc, bufA, bufB, nbr, S(2, 0), S(2, 1), S(2, 2), S(2, 3), S(2, 4),
      S(2, 5), S(2, 6), S(2, 7), S(2, 8), S(2, 9), S(2, 10), 256, 64, 16,
      0.8f * 0.99f, 0);

  // scale 3: 64 -> 1 (global pool to origin), CI=128 EC=128 CO=64 FF=256
  knn_kernel<<<1, 128, 0, stream>>>(pc, nbr, 64, 1, 64, 1);
  scale_kernel<128, 128, 64, 256><<<16, 128, 0, stream>>>(
      pc, bufB, bufA, nbr, S(3, 0), S(3, 1), S(3, 2), S(3, 3), S(3, 4),
      S(3, 5), S(3, 6), S(3, 7), S(3, 8), S(3, 9), S(3, 10), 64, 1, 64,
      3.0f * 0.99f, 1);

  // final per-block RMS norm -> d_out (16 x 9 x 64 fp32)
  final_norm<<<48, 64, 0, stream>>>(bufA, gN, (float*)d_out);
}